// BaseGNNModel_27333171872507
// MI455X (gfx1250) — compile-verified
//
#include <hip/hip_runtime.h>
#include <hip/hip_bf16.h>
#include <cstdint>
#include <cstddef>

// Problem constants (match reference)
#define NB     64
#define NN     20000
#define NE     160000
#define F_TEXT 300
#define HID    1024
#define OUTF   1664
#define NEG_SLOPE 0.2f

typedef __attribute__((ext_vector_type(2))) float v2f;
typedef __attribute__((ext_vector_type(8))) float v8f;

// ---------------------------------------------------------------------------
// Small elementwise / graph kernels
// ---------------------------------------------------------------------------
__global__ void k_zero(float* __restrict__ p, int n) {
  int i = blockIdx.x * blockDim.x + threadIdx.x;
  if (i < n) p[i] = 0.0f;
}

__global__ void k_deg(const int* __restrict__ dst, const float* __restrict__ w,
                      float* __restrict__ deg, int E) {
  int e = blockIdx.x * blockDim.x + threadIdx.x;
  if (e < E) atomicAdd(&deg[dst[e]], w[e]);
}

__global__ void k_dinv(const float* __restrict__ deg, float* __restrict__ dinv, int n) {
  int i = blockIdx.x * blockDim.x + threadIdx.x;
  if (i < n) {
    float d = deg[i];
    dinv[i] = (d > 0.0f) ? rsqrtf(d) : 0.0f;
  }
}

__global__ void k_norm(const int* __restrict__ src, const int* __restrict__ dst,
                       const float* __restrict__ w, const float* __restrict__ dinv,
                       float* __restrict__ nrm, int E) {
  int e = blockIdx.x * blockDim.x + threadIdx.x;
  if (e < E) nrm[e] = dinv[src[e]] * w[e] * dinv[dst[e]];
}

// agg[node][f] = bias[f]  (bias folded into scatter destination init)
__global__ void k_init_bias(float* __restrict__ agg, const float* __restrict__ bias, int F) {
  int f = blockIdx.x * blockDim.x + threadIdx.x;
  int node = blockIdx.y;
  if (f < F) agg[(size_t)node * F + f] = bias[f];
}

__global__ void k_leaky(float* __restrict__ x, int n) {
  int i = blockIdx.x * blockDim.x + threadIdx.x;
  if (i < n) {
    float v = x[i];
    x[i] = (v >= 0.0f) ? v : NEG_SLOPE * v;
  }
}

// One block per edge: agg[dst] += h[src] * norm[e]; agg arrays are L2-resident
// (82/133 MB < 192 MB L2) so float atomics resolve in L2.
__global__ void k_scatter(const float* __restrict__ h, const float* __restrict__ nrm,
                          const int* __restrict__ src, const int* __restrict__ dst,
                          float* __restrict__ agg, int F4) {
  const int e = blockIdx.x;
  const int s = src[e];
  const int d = dst[e];
  const float w = nrm[e];
  const float4* __restrict__ hs = reinterpret_cast<const float4*>(h) + (size_t)s * F4;
  float* __restrict__ ad = agg + (size_t)d * (size_t)(F4 * 4);
  for (int i = threadIdx.x; i < F4; i += blockDim.x) {
    float4 v = hs[i];
    atomicAdd(ad + 4 * i + 0, v.x * w);
    atomicAdd(ad + 4 * i + 1, v.y * w);
    atomicAdd(ad + 4 * i + 2, v.z * w);
    atomicAdd(ad + 4 * i + 3, v.w * w);
  }
}

// ---------------------------------------------------------------------------
// FP32 WMMA GEMM with LDS-staged B panel:  C[M,N] = A[M,K] @ B[K,N]
//
// Block = 256 threads = 8 waves arranged 2 (M) x 4 (N).
// Block tile: 32 rows x 128 cols. Per K-chunk (KC=32) the block stages
// B[kc][128] transposed into LDS as Bt[col][k] (stride 34 floats: even ->
// 8B-aligned ds_load_b64 fragments; 34*n mod 64 distinct over 16 lanes ->
// conflict-free). Each B fragment is then a single LDS b64 read shared by
// the 2 M-waves; global B rows are read once per block, fully coalesced.
//
// Fragment layouts per CDNA5 ISA 7.12.2 (wave32):
//   A 16x4 : row = lane&15, K pair = 2*(lane>>4)
//   B 4x16 : col = lane&15, K pair = 2*(lane>>4)
//   C 16x16: VGPR r -> row = r + 8*(lane>>4), col = lane&15
// Requires M%32==0, N%128==0 (exact for both big GEMMs; K tail handled).
// ---------------------------------------------------------------------------
#define KC 32
#define BT_STRIDE 34

__global__ void k_wmma_gemm_lds(const float* __restrict__ A,
                                const float* __restrict__ B,
                                float* __restrict__ C,
                                int M, int N, int K, int ldb, int ldc) {
  __shared__ float Bt[128 * BT_STRIDE];

  const int tid  = threadIdx.x;
  const int lane = tid & 31;
  const int wave = tid >> 5;
  const int mw   = wave >> 2;              // 0..1
  const int nw   = wave & 3;               // 0..3
  const int mBase = blockIdx.x * 32 + mw * 16;
  const int cBase = blockIdx.y * 128;      // block column base
  const int nBase = cBase + nw * 32;       // wave column base

  const int mr   = lane & 15;
  const int kSel = (lane >> 4) << 1;       // 0 or 2 (even)

  const float* __restrict__ aRow = A + (size_t)(mBase + mr) * K;

  v8f acc0 = {};
  v8f acc1 = {};

  for (int k0 = 0; k0 < K; k0 += KC) {
    const int kc = (K - k0 < KC) ? (K - k0) : KC;

    // Prefetch next chunk of B while we still have this one to chew on.
    if (k0 + KC < K)
      __builtin_prefetch(&B[(size_t)(k0 + KC + (tid >> 7)) * ldb + cBase + (tid & 127)], 0, 1);

    // Cooperative transpose-load: Bt[c][r] = B[k0+r][cBase+c]
    for (int idx = tid; idx < kc * 128; idx += 256) {
      const int r = idx >> 7;
      const int c = idx & 127;
      Bt[c * BT_STRIDE + r] = B[(size_t)(k0 + r) * ldb + cBase + c];
    }
    __syncthreads();

    const float* __restrict__ bt0 = &Bt[(nw * 32 + mr) * BT_STRIDE];
    const float* __restrict__ bt1 = &Bt[(nw * 32 + 16 + mr) * BT_STRIDE];

    for (int kk = 0; kk < kc; kk += 4) {
      const v2f a  = *(const v2f*)(aRow + k0 + kk + kSel);  // global b64 (even offset)
      const v2f b0 = *(const v2f*)(bt0 + kk + kSel);        // ds b64
      const v2f b1 = *(const v2f*)(bt1 + kk + kSel);        // ds b64
      acc0 = __builtin_amdgcn_wmma_f32_16x16x4_f32(false, a, false, b0,
                                                   (short)0, acc0, false, false);
      acc1 = __builtin_amdgcn_wmma_f32_16x16x4_f32(false, a, false, b1,
                                                   (short)0, acc1, false, false);
    }
    __syncthreads();
  }

  const int rBase = (lane >> 4) * 8;
  float* __restrict__ c0 = C + (size_t)(mBase + rBase) * ldc + nBase + mr;
#pragma unroll
  for (int r = 0; r < 8; ++r) {
    c0[(size_t)r * ldc]      = acc0[r];
    c0[(size_t)r * ldc + 16] = acc1[r];
  }
}

// ---------------------------------------------------------------------------
// Direct FP32 WMMA GEMM for C[M,N] = A[M,K] @ B^T where B is [N,K] row-major.
// B fragments are contiguous pairs -> natural global_load_b64. Used for the
// small final GEMM (img_feat @ x2^T, 4.3 GFLOP, x2 L2-resident).
// One wave computes a 16x32 tile. Requires M%16==0, per-wave N tile 32-aligned.
// ---------------------------------------------------------------------------
__global__ void k_wmma_gemm_tn(const float* __restrict__ A,
                               const float* __restrict__ B,
                               float* __restrict__ C,
                               int M, int N, int K, int ldb, int ldc) {
  const int lane  = threadIdx.x & 31;
  const int wave  = threadIdx.x >> 5;
  const int mBase = blockIdx.x * 16;
  const int nBase = (blockIdx.y * (blockDim.x >> 5) + wave) * 32;
  if (mBase >= M || nBase >= N) return;   // wave-uniform: EXEC stays all-1s

  const int mr   = lane & 15;
  const int kSel = (lane >> 4) << 1;

  const float* __restrict__ aRow = A + (size_t)(mBase + mr) * K;
  const float* __restrict__ bRow0 = B + (size_t)(nBase + mr) * ldb;
  const float* __restrict__ bRow1 = B + (size_t)(nBase + 16 + mr) * ldb;

  v8f acc0 = {};
  v8f acc1 = {};

  for (int k0 = 0; k0 < K; k0 += 4) {
    const v2f a  = *(const v2f*)(aRow + k0 + kSel);
    const v2f b0 = *(const v2f*)(bRow0 + k0 + kSel);
    const v2f b1 = *(const v2f*)(bRow1 + k0 + kSel);
    acc0 = __builtin_amdgcn_wmma_f32_16x16x4_f32(false, a, false, b0,
                                                 (short)0, acc0, false, false);
    acc1 = __builtin_amdgcn_wmma_f32_16x16x4_f32(false, a, false, b1,
                                                 (short)0, acc1, false, false);
  }

  const int rBase = (lane >> 4) * 8;
  float* __restrict__ c0 = C + (size_t)(mBase + rBase) * ldc + nBase + mr;
#pragma unroll
  for (int r = 0; r < 8; ++r) {
    c0[(size_t)r * ldc]      = acc0[r];
    c0[(size_t)r * ldc + 16] = acc1[r];
  }
}

// ---------------------------------------------------------------------------
// Launch
// ---------------------------------------------------------------------------
extern "C" void kernel_launch(void* const* d_in, const int* in_sizes, int n_in,
                              void* d_out, int out_size, void* d_ws, size_t ws_size,
                              hipStream_t stream) {
  (void)in_sizes; (void)n_in; (void)out_size; (void)ws_size;

  const float* img_feat      = (const float*)d_in[0];
  const float* node_features = (const float*)d_in[1];
  const int*   edge_src      = (const int*)d_in[2];
  const int*   edge_dst      = (const int*)d_in[3];
  const float* edge_weight   = (const float*)d_in[4];
  const float* W1            = (const float*)d_in[5];
  const float* b1            = (const float*)d_in[6];
  const float* W2            = (const float*)d_in[7];
  const float* b2            = (const float*)d_in[8];
  float* out = (float*)d_out;

  // Workspace carve-out (fp32): deg, dinv, norm, h1, x1, h2, x2  (~432 MB)
  float* ws   = (float*)d_ws;
  float* deg  = ws;  ws += NN;
  float* dinv = ws;  ws += NN;
  float* nrm  = ws;  ws += NE;
  float* h1   = ws;  ws += (size_t)NN * HID;
  float* x1   = ws;  ws += (size_t)NN * HID;
  float* h2   = ws;  ws += (size_t)NN * OUTF;
  float* x2   = ws;  ws += (size_t)NN * OUTF;

  // ---- gcn_norm ----
  k_zero<<<(NN + 255) / 256, 256, 0, stream>>>(deg, NN);
  k_deg<<<(NE + 255) / 256, 256, 0, stream>>>(edge_dst, edge_weight, deg, NE);
  k_dinv<<<(NN + 255) / 256, 256, 0, stream>>>(deg, dinv, NN);
  k_norm<<<(NE + 255) / 256, 256, 0, stream>>>(edge_src, edge_dst, edge_weight,
                                               dinv, nrm, NE);

  // ---- conv1: h1 = X @ W1 ; x1 = leaky(scatter(h1) + b1) ----
  {
    dim3 g(NN / 32, HID / 128);           // 625 x 8, exact coverage
    k_wmma_gemm_lds<<<g, 256, 0, stream>>>(node_features, W1, h1,
                                           NN, HID, F_TEXT, HID, HID);
  }
  k_init_bias<<<dim3((HID + 255) / 256, NN), 256, 0, stream>>>(x1, b1, HID);
  k_scatter<<<NE, 256, 0, stream>>>(h1, nrm, edge_src, edge_dst, x1, HID / 4);
  k_leaky<<<((NN * HID) + 255) / 256, 256, 0, stream>>>(x1, NN * HID);

  // ---- conv2: h2 = x1 @ W2 ; x2 = scatter(h2) + b2 ----
  {
    dim3 g(NN / 32, OUTF / 128);          // 625 x 13, exact coverage
    k_wmma_gemm_lds<<<g, 256, 0, stream>>>(x1, W2, h2,
                                           NN, OUTF, HID, OUTF, OUTF);
  }
  k_init_bias<<<dim3((OUTF + 255) / 256, NN), 256, 0, stream>>>(x2, b2, OUTF);
  k_scatter<<<NE, 256, 0, stream>>>(h2, nrm, edge_src, edge_dst, x2, OUTF / 4);

  // ---- out = img_feat @ x2^T : [64,1664] x [1664,20000] ----
  {
    dim3 g(NB / 16, (NN + 255) / 256);    // 4 x 79 (20000 % 32 == 0, no partials)
    k_wmma_gemm_tn<<<g, 256, 0, stream>>>(img_feat, x2, out,
                                          NB, NN, OUTF, OUTF, NN);
  }
}